// GATDecoder_84284438217360
// MI455X (gfx1250) — compile-verified
//
#include <hip/hip_runtime.h>
#include <math.h>

#define NEG_SLOPE 0.2f
#define EPS_BN 1e-5f
#define HEADS 8
#define KDIM 16
#define CH 128
#define NODE_NUM 10000
#define BGR 8

typedef float v2f __attribute__((ext_vector_type(2)));
typedef float v8f __attribute__((ext_vector_type(8)));
typedef unsigned u32x4 __attribute__((ext_vector_type(4)));
typedef unsigned u32x8 __attribute__((ext_vector_type(8)));

__device__ __forceinline__ float lrelu(float x) { return x >= 0.f ? x : NEG_SLOPE * x; }

// monotonic float<->uint encoding for atomicMax-based segment max
__device__ __forceinline__ unsigned fenc(float f) {
    unsigned b = __float_as_uint(f);
    return (b & 0x80000000u) ? ~b : (b | 0x80000000u);
}
__device__ __forceinline__ float fdec(unsigned u) {
    return (u & 0x80000000u) ? __uint_as_float(u & 0x7FFFFFFFu) : __uint_as_float(~u);
}

__global__ void fill_zero_k(float* __restrict__ p, long n) {
    long i = (long)blockIdx.x * blockDim.x + threadIdx.x;
    if (i < n) p[i] = 0.f;
}

// H[M x 128] = X[M x K] * W^T, W is [128 x K] row-major.
// One block: 16 rows of X staged into LDS by the Tensor Data Mover (2D tile DMA
// with LDS padding = K+4 row stride to dodge the 64-bank conflicts), then 8
// waves each own a 16-wide N tile on the fp32 WMMA path
// (V_WMMA_F32_16X16X4_F32, wave32).
template <int K>
__global__ __launch_bounds__(256) void wmma_gemm_k(const float* __restrict__ X,
                                                   const float* __restrict__ W,
                                                   float* __restrict__ H) {
    constexpr int KP = K + 4;                              // padded LDS row stride
    constexpr unsigned PAD_INTERVAL = (K == 8) ? 2u : 6u;  // pad after 8 / 128 DWORDs
    constexpr unsigned PAD_AMOUNT = 3u;                    // pad 4 DWORDs
    __shared__ float Xs[16 * KP];
    const int m0 = blockIdx.x * 16;

    if (threadIdx.y == 0) {
        // Build Tensor DMA Descriptor (D#): 2D tile, 16 rows x K fp32, stride K.
        unsigned long long ga = (unsigned long long)(const void*)(X + (long)m0 * K);
        unsigned lds = (unsigned)(unsigned long long)(void*)Xs;  // low 32 bits = LDS offset
        u32x4 g0;
        g0[0] = 1u;                                      // count=1, user descriptor
        g0[1] = lds;                                     // lds_addr
        g0[2] = (unsigned)ga;                            // global_addr[31:0]
        g0[3] = (unsigned)(ga >> 32) | 0x80000000u;      // global_addr[56:32], type=2
        u32x8 g1;
        g1[0] = 0x00020000u | (1u << 20) | (PAD_INTERVAL << 22) | (PAD_AMOUNT << 25);
        //        data_size=4B   pad_enable     pad_interval         pad_amount
        g1[1] = ((unsigned)K) << 16;                     // tensor_dim0 lo16 = K
        g1[2] = 16u << 16;                               // tensor_dim1 lo16 = 16
        g1[3] = ((unsigned)K) << 16;                     // tile_dim0 = K
        g1[4] = 16u;                                     // tile_dim1 = 16 (tile_dim2 = 0)
        g1[5] = (unsigned)K;                             // tensor_dim0_stride lo32 = K
        g1[6] = 0u;
        g1[7] = 0u;
        asm volatile("tensor_load_to_lds %0, %1" :: "s"(g0), "s"(g1) : "memory");
        __builtin_amdgcn_s_wait_tensorcnt(0);
    }
    __syncthreads();

    const int lane = threadIdx.x;
    const int half = lane >> 4;      // 0: K pair {k0,k0+1}, 1: {k0+2,k0+3}
    const int lr   = lane & 15;      // M index for A, N index for B
    const int n0   = threadIdx.y * 16;

    v8f acc = {};
    for (int k0 = 0; k0 < K; k0 += 4) {
        int kk = k0 + 2 * half;
        v2f a = *(const v2f*)&Xs[lr * KP + kk];                 // A: 16x4 slab of X
        v2f b = *(const v2f*)&W[(long)(n0 + lr) * K + kk];      // B: 4x16 slab of W^T
        acc = __builtin_amdgcn_wmma_f32_16x16x4_f32(
            false, a, false, b, (short)0, acc, false, false);
    }
#pragma unroll
    for (int r = 0; r < 8; ++r) {
        // C layout: VGPR r -> M = r (lanes 0-15) / r+8 (lanes 16-31), N = lane%16
        H[(long)(m0 + r + 8 * half) * CH + n0 + lr] = acc[r];
    }
}

// per-node attention partials: sdst[n,h] = <h[n,h,:], att[h,:16]>, ssrc uses att[h,16:]
__global__ void node_att_k(const float* __restrict__ H, const float* __restrict__ att,
                           float* __restrict__ sdst, float* __restrict__ ssrc, int N) {
    int i = blockIdx.x * blockDim.x + threadIdx.x;
    if (i >= N * HEADS) return;
    int n = i >> 3, h = i & 7;
    const float* hp = H + (long)n * CH + h * KDIM;
    const float* ad = att + h * 2 * KDIM;
    float s0 = 0.f, s1 = 0.f;
#pragma unroll
    for (int k = 0; k < KDIM; ++k) {
        float v = hp[k];
        s0 += v * ad[k];
        s1 += v * ad[KDIM + k];
    }
    sdst[i] = s0;
    ssrc[i] = s1;
}

__device__ __forceinline__ void edge_sd(const int* __restrict__ ei, int E, int e,
                                        int& s, int& d) {
    if (e < E) { s = ei[e]; d = ei[E + e]; }
    else       { s = d = e - E; }   // appended self-loops
}

__global__ void edge_amax_k(const int* __restrict__ ei, int E, int Etot,
                            const float* __restrict__ sdst, const float* __restrict__ ssrc,
                            unsigned* __restrict__ amax) {
    int i = blockIdx.x * blockDim.x + threadIdx.x;
    if (i >= Etot * HEADS) return;
    int e = i >> 3, h = i & 7;
    int s, d; edge_sd(ei, E, e, s, d);
    float a = lrelu(sdst[d * HEADS + h] + ssrc[s * HEADS + h]);
    atomicMax(&amax[s * HEADS + h], fenc(a));
}

__global__ void edge_den_k(const int* __restrict__ ei, int E, int Etot,
                           const float* __restrict__ sdst, const float* __restrict__ ssrc,
                           const unsigned* __restrict__ amax, float* __restrict__ den) {
    int i = blockIdx.x * blockDim.x + threadIdx.x;
    if (i >= Etot * HEADS) return;
    int e = i >> 3, h = i & 7;
    int s, d; edge_sd(ei, E, e, s, d);
    float a = lrelu(sdst[d * HEADS + h] + ssrc[s * HEADS + h]);
    float t = expf(a - fdec(amax[s * HEADS + h]));
    atomicAdd(&den[s * HEADS + h], t);
}

__global__ void edge_scatter_k(const int* __restrict__ ei, int E, int Etot,
                               const float* __restrict__ sdst, const float* __restrict__ ssrc,
                               const unsigned* __restrict__ amax, const float* __restrict__ den,
                               const float* __restrict__ H, float* __restrict__ agg) {
    int i = blockIdx.x * blockDim.x + threadIdx.x;
    if (i >= Etot * HEADS) return;
    int e = i >> 3, h = i & 7;
    int s, d; edge_sd(ei, E, e, s, d);
    const float* hs = H + (long)s * CH + h * KDIM;
    __builtin_prefetch(hs, 0, 0);      // global_prefetch_b8: hide random-gather latency
    float a = lrelu(sdst[d * HEADS + h] + ssrc[s * HEADS + h]);
    float w = expf(a - fdec(amax[s * HEADS + h])) / (den[s * HEADS + h] + 1e-16f);
    float* op = agg + (long)d * CH + h * KDIM;
#pragma unroll
    for (int k = 0; k < KDIM; ++k) atomicAdd(&op[k], hs[k] * w);
}

// y = relu(agg + bias) in place, accumulate per-channel sum / sumsq
__global__ void stats_bias_relu_k(float* __restrict__ y, const float* __restrict__ bias,
                                  float* __restrict__ stats, int N) {
    const int ROWS = 80;
    int c = threadIdx.x;          // 0..127
    int r0 = blockIdx.x * ROWS;
    float b = bias[c];
    float s = 0.f, q = 0.f;
    for (int r = 0; r < ROWS; ++r) {
        int row = r0 + r;
        if (row >= N) break;
        long idx = (long)row * CH + c;
        float v = y[idx] + b;
        v = v > 0.f ? v : 0.f;
        y[idx] = v;
        s += v;
        q += v * v;
    }
    atomicAdd(&stats[c], s);
    atomicAdd(&stats[CH + c], q);
}

__global__ void bn_lrelu_k(float* __restrict__ y, const float* __restrict__ stats,
                           const float* __restrict__ g, const float* __restrict__ be, int N) {
    long i = (long)blockIdx.x * blockDim.x + threadIdx.x;
    if (i >= (long)N * CH) return;
    int c = (int)(i & (CH - 1));
    float m = stats[c] / (float)N;
    float var = stats[CH + c] / (float)N - m * m;
    float v = (y[i] - m) * rsqrtf(var + EPS_BN) * g[c] + be[c];
    y[i] = lrelu(v);
}

__global__ void powermean_k(const float* __restrict__ x, const float* __restrict__ pp,
                            float* __restrict__ xg) {
    int b = blockIdx.x, c = threadIdx.x;
    float p = *pp;
    float acc = 0.f;
    const float* base = x + (long)b * NODE_NUM * CH + c;
    for (int n = 0; n < NODE_NUM; ++n) {
        float v = base[(long)n * CH];
        v = fminf(fmaxf(v, 0.f), 100.f);
        acc += powf(v, p);
    }
    acc /= (float)NODE_NUM;
    acc = fminf(fmaxf(acc, 0.f), 100.f);
    xg[b * CH + c] = powf(acc, 1.f / p);
}

__global__ void head_k(const float* __restrict__ xg, const float* __restrict__ Wg,
                       const float* __restrict__ bg, float* __restrict__ out) {
    __shared__ float o[BGR * 2];
    int t = threadIdx.x;
    if (t < BGR * 2) {
        int b = t >> 1, j = t & 1;
        float acc = bg[j];
        for (int c = 0; c < CH; ++c) acc += xg[b * CH + c] * Wg[j * CH + c];
        o[t] = acc;
        out[t] = acc;
    }
    __syncthreads();
    if (t < BGR) out[BGR * 2 + t] = (o[2 * t + 1] > o[2 * t]) ? 1.f : 0.f;  // argmax
}

extern "C" void kernel_launch(void* const* d_in, const int* in_sizes, int n_in,
                              void* d_out, int out_size, void* d_ws, size_t ws_size,
                              hipStream_t stream) {
    const float* x   = (const float*)d_in[0];
    const int*   ei  = (const int*)d_in[1];
    const float* W1  = (const float*)d_in[2];
    const float* att1= (const float*)d_in[3];
    const float* b1  = (const float*)d_in[4];
    const float* g1  = (const float*)d_in[5];
    const float* be1 = (const float*)d_in[6];
    const float* W2  = (const float*)d_in[7];
    const float* att2= (const float*)d_in[8];
    const float* b2  = (const float*)d_in[9];
    const float* g2  = (const float*)d_in[10];
    const float* be2 = (const float*)d_in[11];
    const float* W3  = (const float*)d_in[12];
    const float* att3= (const float*)d_in[13];
    const float* b3  = (const float*)d_in[14];
    const float* g3  = (const float*)d_in[15];
    const float* be3 = (const float*)d_in[16];
    const float* p   = (const float*)d_in[17];
    const float* Wg  = (const float*)d_in[18];
    const float* bg  = (const float*)d_in[19];
    float* out = (float*)d_out;

    const int N = in_sizes[0] / 8;       // 80000
    const int E = in_sizes[1] / 2;       // 1,280,000
    const int Etot = E + N;              // with self-loops

    // workspace carve-up
    char* ws = (char*)d_ws;
    size_t off = 0;
    auto carve = [&](size_t bytes) -> void* {
        void* q = ws + off;
        off = (off + bytes + 255) & ~(size_t)255;
        return q;
    };
    float*    H     = (float*)carve((size_t)N * CH * 4);
    float*    FA    = (float*)carve((size_t)N * CH * 4);
    float*    FB    = (float*)carve((size_t)N * CH * 4);
    float*    sdst  = (float*)carve((size_t)N * HEADS * 4);
    float*    ssrc  = (float*)carve((size_t)N * HEADS * 4);
    unsigned* amax  = (unsigned*)carve((size_t)N * HEADS * 4);
    float*    den   = (float*)carve((size_t)N * HEADS * 4);
    float*    stats = (float*)carve(2 * CH * 4);
    float*    xg    = (float*)carve(BGR * CH * 4);

    const long nfeat = (long)N * CH;
    const int  gFeat = (int)((nfeat + 255) / 256);
    const int  gNH   = (N * HEADS + 255) / 256;
    const int  gEH   = (int)(((long)Etot * HEADS + 255) / 256);
    const int  gBN   = (N + 79) / 80;

    auto layer = [&](const float* xin, int K, const float* W, const float* att,
                     const float* bias, const float* g, const float* be, float* agg) {
        fill_zero_k<<<gFeat, 256, 0, stream>>>(agg, nfeat);
        fill_zero_k<<<gNH, 256, 0, stream>>>((float*)amax, (long)N * HEADS);
        fill_zero_k<<<gNH, 256, 0, stream>>>(den, (long)N * HEADS);
        fill_zero_k<<<1, 256, 0, stream>>>(stats, 2 * CH);

        if (K == 8)
            wmma_gemm_k<8><<<N / 16, dim3(32, 8), 0, stream>>>(xin, W, H);
        else
            wmma_gemm_k<128><<<N / 16, dim3(32, 8), 0, stream>>>(xin, W, H);

        node_att_k<<<gNH, 256, 0, stream>>>(H, att, sdst, ssrc, N);
        edge_amax_k<<<gEH, 256, 0, stream>>>(ei, E, Etot, sdst, ssrc, amax);
        edge_den_k<<<gEH, 256, 0, stream>>>(ei, E, Etot, sdst, ssrc, amax, den);
        edge_scatter_k<<<gEH, 256, 0, stream>>>(ei, E, Etot, sdst, ssrc, amax, den, H, agg);
        stats_bias_relu_k<<<gBN, 128, 0, stream>>>(agg, bias, stats, N);
        bn_lrelu_k<<<gFeat, 256, 0, stream>>>(agg, stats, g, be, N);
    };

    layer(x,  8,   W1, att1, b1, g1, be1, FA);
    layer(FA, 128, W2, att2, b2, g2, be2, FB);
    layer(FB, 128, W3, att3, b3, g3, be3, FA);

    powermean_k<<<BGR, CH, 0, stream>>>(FA, p, xg);
    head_k<<<1, 32, 0, stream>>>(xg, Wg, bg, out);
}